// RadianceRenderer_44787918963337
// MI455X (gfx1250) — compile-verified
//
#include <hip/hip_runtime.h>

typedef __attribute__((ext_vector_type(2))) float v2f;
typedef __attribute__((ext_vector_type(8))) float v8f;

// Swap the two 16-lane halves of the wave32 entirely in the VALU
// (v_permlanex16_b32 with identity lane-selects == xor-16 shuffle, no DS).
static __device__ __forceinline__ float xor16(float v) {
    int i = __builtin_bit_cast(int, v);
    int r = __builtin_amdgcn_permlanex16(i, i, 0x76543210u, 0xfedcba98u,
                                         false, false);
    return __builtin_bit_cast(float, r);
}

static __device__ __forceinline__ v8f wmma4(v2f a, v2f b, v8f c) {
    // D = A(16x4,f32) x B(4x16,f32) + C(16x16,f32), exact f32 XDL path
    return __builtin_amdgcn_wmma_f32_16x16x4_f32(false, a, false, b, (short)0, c,
                                                 false, false);
}

__global__ __launch_bounds__(256) void nerf_render_kernel(
    const float* __restrict__ rays_o, const float* __restrict__ rays_d,
    const float* __restrict__ U,      const float* __restrict__ aabb,
    const float* __restrict__ W1,     const float* __restrict__ b1,
    const float* __restrict__ Wsig,   const float* __restrict__ Wcol,
    const float* __restrict__ Wdir,   float* __restrict__ out,
    int N, int T)
{
    const int  lane = threadIdx.x & 31;
    const int  ray  = blockIdx.x * blockDim.x + threadIdx.x;
    const int  rc   = (ray < N) ? ray : (N - 1);   // clamp loads, keep EXEC full
    const int  m    = lane & 15;
    const bool hiL  = lane >= 16;

    // ---- constant operands in WMMA register layouts (built once per wave) ----
    // A1 = W1^T in 16x4 A-layout; K3 column is ZERO (bias goes in via C).
    v2f A1;
    A1.x = hiL ? W1[2 * 16 + m] : W1[0 * 16 + m];
    A1.y = hiL ? 0.0f           : W1[1 * 16 + m];

    // Bias as the C accumulator of the layer-1 WMMA (C-layout: vgpr r holds
    // row r in lanes 0..15 and row r+8 in lanes 16..31, bcast over columns).
    v8f Cbias;
    #pragma unroll
    for (int r = 0; r < 8; ++r) Cbias[r] = hiL ? b1[r + 8] : b1[r];

    // W2 = [Wsigma | Wcolor] (16 hidden x 4 channels).
    // Layer-2 chunk k uses K-set {2k, 2k+1, 2k+8, 2k+9}: with this ordering the
    // B operand is exactly the layer-1 D register pair (H[2k],H[2k+1]) -- the
    // D-layout (vgpr r = rows r / r+8 across lane halves) IS the B-layout.
    // A2lo: channels in rows 0..3  -> tile0 result lands in lanes 0..15.
    // A2hi: channels in rows 8..11 -> tile1 result lands in lanes 16..31,
    //       directly in the lane that owns the ray (no output merge shuffle).
    auto w2 = [&](int j, int c) -> float {
        return (c == 0) ? Wsig[j] : Wcol[j * 3 + (c - 1)];
    };
    v2f A2lo[4], A2hi[4];
    #pragma unroll
    for (int k = 0; k < 4; ++k) {
        const int j0 = 2 * k + (hiL ? 8 : 0);   // K0 (lo half) / K2 (hi half)
        float l0v = 0.f, l1v = 0.f, h0v = 0.f, h1v = 0.f;
        if (m < 4) {                 // tile0: channel = m, rows 0..3
            l0v = w2(j0, m);
            l1v = w2(j0 + 1, m);
        }
        if (m >= 8 && m < 12) {      // tile1: channel = m-8, rows 8..11
            h0v = w2(j0, m - 8);
            h1v = w2(j0 + 1, m - 8);
        }
        A2lo[k].x = l0v;  A2lo[k].y = l1v;
        A2hi[k].x = h0v;  A2hi[k].y = h1v;
    }

    // ---- per-ray scalar setup: slab intersection, dir encoding ----
    const float ox = rays_o[3 * rc + 0], oy = rays_o[3 * rc + 1], oz = rays_o[3 * rc + 2];
    const float dx = rays_d[3 * rc + 0], dy = rays_d[3 * rc + 1], dz = rays_d[3 * rc + 2];
    const float eps = 1e-9f;
    const float sdx = (fabsf(dx) < eps) ? eps : dx;
    const float sdy = (fabsf(dy) < eps) ? eps : dy;
    const float sdz = (fabsf(dz) < eps) ? eps : dz;
    const float bl0 = aabb[0], bl1 = aabb[1], bl2 = aabb[2];
    const float bh0 = aabb[3], bh1 = aabb[4], bh2 = aabb[5];
    const float t1x = (bl0 - ox) / sdx, t2x = (bh0 - ox) / sdx;
    const float t1y = (bl1 - oy) / sdy, t2y = (bh1 - oy) / sdy;
    const float t1z = (bl2 - oz) / sdz, t2z = (bh2 - oz) / sdz;
    float tn = fmaxf(fmaxf(fminf(t1x, t2x), fminf(t1y, t2y)), fminf(t1z, t2z));
    tn = fmaxf(tn, 0.0f);
    const float tf = fminf(fminf(fmaxf(t1x, t2x), fmaxf(t1y, t2y)), fmaxf(t1z, t2z));
    const float active = (tn < tf) ? 1.0f : 0.0f;

    const float dnorm  = sqrtf(dx * dx + dy * dy + dz * dz);
    const float idn    = 1.0f / dnorm;
    const float uxn = dx * idn, uyn = dy * idn, uzn = dz * idn;
    const float enc0 = uxn * Wdir[0] + uyn * Wdir[3] + uzn * Wdir[6];
    const float enc1 = uxn * Wdir[1] + uyn * Wdir[4] + uzn * Wdir[7];
    const float enc2 = uxn * Wdir[2] + uyn * Wdir[5] + uzn * Wdir[8];

    const float tfinal = tf * 10.0f;                    // RAY_EXT_FACTOR
    const float stepw  = (tf - tn) * (1.0f / (float)T); // (tfar-tnear)/T

    // One full MLP sample at step t -> (sigma, r, g, b, ts) for this lane's ray
    auto sample = [&](int t, float& o_sig, float& o_r, float& o_g, float& o_b,
                      float& o_ts) {
        const float uu = U[(size_t)t * (size_t)N + (size_t)rc];
        const float ts = fmaf((float)t + uu, stepw, tn);
        const float px = fmaf(ts, dx, ox);
        const float py = fmaf(ts, dy, oy);
        const float pz = fmaf(ts, dz, oz);

        // X^T as 4x16 B-layout (rows x,y | z,dc); one per 16-ray tile.
        // K3 column of A1 is zero, so B.y in the "other" half is don't-care:
        // only 3 cross-half moves and 2 selects per sample.
        const float xs = xor16(px), ys = xor16(py), zs = xor16(pz);
        v2f B0, B1;
        B0.x = hiL ? zs : px;  B0.y = py;   // tile0: rays lanes 0..15
        B1.x = hiL ? pz : xs;  B1.y = ys;   // tile1: rays lanes 16..31

        v8f H0 = wmma4(A1, B0, Cbias);   // H^T = W1^T X^T + b1 (rows=hidden)
        v8f H1 = wmma4(A1, B1, Cbias);
        #pragma unroll
        for (int i = 0; i < 8; ++i) {
            H0[i] = fmaxf(H0[i], 0.0f);
            H1[i] = fmaxf(H1[i], 0.0f);
        }

        // Layer 2: 4 chained K=4 WMMAs per tile; B operands are the H register
        // pairs directly (K-order {2k,2k+1,2k+8,2k+9}) -- no shuffles needed.
        v2f b00 = {H0[0], H0[1]}, b01 = {H0[2], H0[3]};
        v2f b02 = {H0[4], H0[5]}, b03 = {H0[6], H0[7]};
        v2f b10 = {H1[0], H1[1]}, b11 = {H1[2], H1[3]};
        v2f b12 = {H1[4], H1[5]}, b13 = {H1[6], H1[7]};
        v8f O0 = {}, O1 = {};
        O0 = wmma4(A2lo[3], b03, O0);
        O1 = wmma4(A2hi[3], b13, O1);
        O0 = wmma4(A2lo[2], b02, O0);
        O1 = wmma4(A2hi[2], b12, O1);
        O0 = wmma4(A2lo[1], b01, O0);
        O1 = wmma4(A2hi[1], b11, O1);
        O0 = wmma4(A2lo[0], b00, O0);
        O1 = wmma4(A2hi[0], b10, O1);

        // tile0 channels sit in lanes 0..15 (rows 0..3), tile1 channels in
        // lanes 16..31 (rows 8..11) -> each lane already owns its ray's values.
        const float v0 = hiL ? O1[0] : O0[0];
        const float v1 = hiL ? O1[1] : O0[1];
        const float v2 = hiL ? O1[2] : O0[2];
        const float v3 = hiL ? O1[3] : O0[3];

        // sigma = softplus(v0); color = sigmoid(v + dir_enc)
        o_sig = fmaxf(v0, 0.0f) + __logf(1.0f + __expf(-fabsf(v0)));
        o_r   = __fdividef(1.0f, 1.0f + __expf(-(v1 + enc0)));
        o_g   = __fdividef(1.0f, 1.0f + __expf(-(v2 + enc1)));
        o_b   = __fdividef(1.0f, 1.0f + __expf(-(v3 + enc2)));
        o_ts  = ts;
    };

    // ---- streaming transmittance integration (exclusive-cumsum form) ----
    float Ttr = 1.f, amap = 0.f, dmap = 0.f, cr = 0.f, cg = 0.f, cb = 0.f;
    float p_sig, p_r, p_g, p_b, p_ts;
    sample(0, p_sig, p_r, p_g, p_b, p_ts);   // peeled first sample

    for (int t = 1; t < T; ++t) {
        float sig, r, g, b, ts;
        sample(t, sig, r, g, b, ts);

        const float sd = p_sig * (ts - p_ts) * dnorm;
        const float es = __expf(-sd);
        const float w  = (1.0f - es) * Ttr;
        Ttr *= es;
        cr += w * p_r;  cg += w * p_g;  cb += w * p_b;
        amap += w;      dmap += w * p_ts;

        p_sig = sig; p_r = r; p_g = g; p_b = b; p_ts = ts;
    }
    {   // last sample: delta = tfar*10 - ts[T-1]
        const float sd = p_sig * (tfinal - p_ts) * dnorm;
        const float es = __expf(-sd);
        const float w  = (1.0f - es) * Ttr;
        cr += w * p_r;  cg += w * p_g;  cb += w * p_b;
        amap += w;      dmap += w * p_ts;
    }

    if (ray < N) {
        float* o5 = out + (size_t)ray * 5;
        o5[0] = cr   * active;
        o5[1] = cg   * active;
        o5[2] = cb   * active;
        o5[3] = amap * active;
        o5[4] = dmap * active;
    }
}

extern "C" void kernel_launch(void* const* d_in, const int* in_sizes, int n_in,
                              void* d_out, int out_size, void* d_ws, size_t ws_size,
                              hipStream_t stream) {
    const float* rays_o = (const float*)d_in[0];
    const float* rays_d = (const float*)d_in[1];
    const float* U      = (const float*)d_in[2];
    const float* aabb   = (const float*)d_in[3];
    const float* W1     = (const float*)d_in[4];
    const float* b1     = (const float*)d_in[5];
    const float* Wsig   = (const float*)d_in[6];
    const float* Wcol   = (const float*)d_in[7];
    const float* Wdir   = (const float*)d_in[8];
    float* out = (float*)d_out;

    const int N = in_sizes[0] / 3;
    const int T = (N > 0) ? (in_sizes[2] / N) : 0;
    const int block = 256;                 // 8 wave32s per WGP launch unit
    const int grid  = (N + block - 1) / block;
    nerf_render_kernel<<<grid, block, 0, stream>>>(rays_o, rays_d, U, aabb,
                                                   W1, b1, Wsig, Wcol, Wdir,
                                                   out, N, T);
}